// TopLoss_72919954751829
// MI455X (gfx1250) — compile-verified
//
#include <hip/hip_runtime.h>

// CDNA5 / gfx1250, wave32. Single-workgroup latency-bound kernel:
//   - 256 threads, one per (b,c,d) cell; per-cell Frobenius norm in f32.
//   - Final 256-element reduction via V_WMMA_F32_16X16X4_F32 with an all-ones
//     A matrix: D[m][n] = sum_k B[k][n], so the lane-sum of accumulator VGPR0
//     equals 2 * (sum of all B elements) regardless of the exact B register
//     layout (permutation-invariant). 4 accumulating WMMAs cover 256 values.

typedef __attribute__((ext_vector_type(2))) float v2f;
typedef __attribute__((ext_vector_type(8))) float v8f;

#define CELLS 256   // B*C*D = 32*4*2
#define NP    48
#define NG    32

__global__ __launch_bounds__(256) void toploss_kernel(
    const float* __restrict__ pred,     // [256, 48, 2]
    const float* __restrict__ ground,   // [256, 32, 2]
    const int*   __restrict__ midx,     // [256, 48]
    float*       __restrict__ out)      // [1]
{
    __shared__ float s_norm[CELLS];

    const int t = threadIdx.x;                       // cell id 0..255
    const float2* p2 = reinterpret_cast<const float2*>(pred)   + t * NP;
    const float2* g2 = reinterpret_cast<const float2*>(ground) + t * NG;
    const int*    mi = midx + t * NP;

    float acc = 0.0f;
#pragma unroll 4
    for (int i = 0; i < NP; ++i) {
        float2 p = p2[i];                            // global_load_b64
        int    m = mi[i];
        float  t0, t1;
        if (m >= 0) {
            float2 g = g2[m];
            t0 = g.x; t1 = g.y;
        } else {
            float pr = 0.5f * (p.x + p.y);           // diagonal projection
            t0 = pr;  t1 = pr;
        }
        float d0 = p.x - t0;
        float d1 = p.y - t1;
        acc = fmaf(d0, d0, acc);
        acc = fmaf(d1, d1, acc);
    }
    s_norm[t] = sqrtf(acc);                          // per-cell Frobenius norm
    __syncthreads();

    // ---- WMMA reduction of the 256 norms (all 8 waves do it redundantly so
    // EXEC is all-ones for every wave issuing a WMMA; ISA requirement). ----
    const int lane = t & 31;

    v2f a;                                           // A = ones(16x4): 2 VGPRs
    a[0] = 1.0f; a[1] = 1.0f;

    v8f c = {0.f, 0.f, 0.f, 0.f, 0.f, 0.f, 0.f, 0.f};

#pragma unroll
    for (int q = 0; q < 4; ++q) {
        v2f b;                                       // B slots: 64 norms/WMMA
        b[0] = s_norm[q * 64 + lane];
        b[1] = s_norm[q * 64 + 32 + lane];
        // D = ones(16x4) x B(4x16) + C  -> every row of D = column sums of B
        c = __builtin_amdgcn_wmma_f32_16x16x4_f32(
                /*neg_a=*/false, a, /*neg_b=*/false, b,
                /*c_mod=*/(short)0, c, /*reuse_a=*/false, /*reuse_b=*/false);
    }

    // VGPR0 across 32 lanes holds column sums twice (rows M=0 and M=8),
    // so the full-lane sum equals 2 * sum(all 256 norms).
    float c0 = c[0];
#pragma unroll
    for (int off = 16; off > 0; off >>= 1)
        c0 += __shfl_xor(c0, off, 32);

    // mean over (b,c) of sum over d  ==  total / 128; extra 1/2 undoes the
    // row duplication above -> divide by 256.
    if (t == 0)
        out[0] = c0 * (1.0f / 256.0f);
}

extern "C" void kernel_launch(void* const* d_in, const int* in_sizes, int n_in,
                              void* d_out, int out_size, void* d_ws, size_t ws_size,
                              hipStream_t stream)
{
    const float* pred   = (const float*)d_in[0];   // [32,4,2,48,2] f32
    const float* ground = (const float*)d_in[1];   // [32,4,2,32,2] f32
    const int*   midx   = (const int*)  d_in[2];   // [32,4,2,48]   i32
    float*       out    = (float*)d_out;           // scalar

    toploss_kernel<<<1, 256, 0, stream>>>(pred, ground, midx, out);
}